// Z_88871463288920
// MI455X (gfx1250) — compile-verified
//
#include <hip/hip_runtime.h>

typedef __attribute__((ext_vector_type(2))) float v2f;
typedef __attribute__((ext_vector_type(8))) float v8f;

#define ZN    8192
#define ZHALF 4096

#if __has_builtin(__builtin_amdgcn_wmma_f32_16x16x4_f32)
#define Z_HAVE_WMMA_F32 1
#else
#define Z_HAVE_WMMA_F32 0
#endif

#if Z_HAVE_WMMA_F32
// One wave handles one 16-row tile = 16 elements of half 0 + 16 of half 1.
// 256 tiles total -> 32 blocks x 256 threads (8 waves/block).
__global__ __launch_bounds__(256) void zgate_wmma_kernel(
    const float* __restrict__ xr, const float* __restrict__ xi,
    const float* __restrict__ mr, const float* __restrict__ mi,
    float* __restrict__ out) {
  const int lane = threadIdx.x & 31;
  const int wave = threadIdx.x >> 5;
  const int tile = blockIdx.x * 8 + wave;   // 0..255
  const int base = tile * 16;               // within-half offset, 0..4080

  // ---- A matrix (16x4 f32): row M = [xr0, xi0, xr1, xi1] ----
  // Documented layout: VGPR0 -> K=0 (lanes 0-15) / K=2 (lanes 16-31),
  //                    VGPR1 -> K=1 (lanes 0-15) / K=3 (lanes 16-31).
  // => lanes 0-15 load half-0 elements, lanes 16-31 load half-1 elements.
  const int col = lane & 15;
  const bool lo = lane < 16;
  const int idx = lo ? (base + col) : (ZHALF + base + col);
  v2f a;
  a.x = xr[idx];  // K = 0 or 2
  a.y = xi[idx];  // K = 1 or 3

  // ---- B matrix (4x16 f32): the two 2x2 rotation blocks ----
  // B[:,0]=[mr0,-mi0,0,0]  B[:,1]=[mi0,mr0,0,0]
  // B[:,2]=[0,0,mr1,-mi1]  B[:,3]=[0,0,mi1,mr1]   (other columns zero)
  // Assumed striping mirrors A/C/D: VGPR0 -> rows K=0 / K=2, VGPR1 -> K=1 / K=3.
  const float mr0 = mr[0], mi0 = mi[0], mr1 = mr[1], mi1 = mi[1];
  const float bx_lo = (col == 0) ?  mr0 : (col == 1) ? mi0 : 0.0f; // B[0,col]
  const float by_lo = (col == 0) ? -mi0 : (col == 1) ? mr0 : 0.0f; // B[1,col]
  const float bx_hi = (col == 2) ?  mr1 : (col == 3) ? mi1 : 0.0f; // B[2,col]
  const float by_hi = (col == 2) ? -mi1 : (col == 3) ? mr1 : 0.0f; // B[3,col]
  v2f b;
  b.x = lo ? bx_lo : bx_hi;
  b.y = lo ? by_lo : by_hi;

  // ---- D = A x B (+0): D[M,0..3] = (yr0, yi0, yr1, yi1) for row M ----
  v8f d = {};
  d = __builtin_amdgcn_wmma_f32_16x16x4_f32(
      /*neg_a=*/false, a, /*neg_b=*/false, b,
      /*c_mod=*/(short)0, d, /*reuse_a=*/false, /*reuse_b=*/false);

  // ---- Store: D VGPR v holds M=v (lanes 0-15) / M=v+8 (lanes 16-31) ----
  // col 0 -> Re(half0), col 1 -> Im(half0), col 2 -> Re(half1), col 3 -> Im(half1)
  const int rowadd = lo ? 0 : 8;
  if (col < 4) {
    const int half = col >> 1;  // which half of the state
    const int reim = col & 1;   // 0 = real plane, 1 = imag plane
#pragma unroll
    for (int v = 0; v < 8; ++v) {
      const int row = v + rowadd;
      const int elem = half * ZHALF + base + row;
      out[reim * ZN + elem] = d[v];
    }
  }
}
#else
// Fallback: plain elementwise complex phase multiply (one thread per element).
__global__ __launch_bounds__(256) void zgate_valu_kernel(
    const float* __restrict__ xr, const float* __restrict__ xi,
    const float* __restrict__ mr, const float* __restrict__ mi,
    float* __restrict__ out) {
  const int t = blockIdx.x * blockDim.x + threadIdx.x;
  if (t >= ZN) return;
  const int d = t >> 12;  // wire-0 digit (most significant axis)
  const float xrv = xr[t], xiv = xi[t];
  const float mrv = mr[d], miv = mi[d];
  out[t]      = xrv * mrv - xiv * miv;
  out[ZN + t] = xrv * miv + xiv * mrv;
}
#endif

extern "C" void kernel_launch(void* const* d_in, const int* in_sizes, int n_in,
                              void* d_out, int out_size, void* d_ws, size_t ws_size,
                              hipStream_t stream) {
  (void)in_sizes; (void)n_in; (void)out_size; (void)d_ws; (void)ws_size;
  const float* xr = (const float*)d_in[0];
  const float* xi = (const float*)d_in[1];
  const float* mr = (const float*)d_in[2];
  const float* mi = (const float*)d_in[3];
  float* out = (float*)d_out;
#if Z_HAVE_WMMA_F32
  zgate_wmma_kernel<<<dim3(32), dim3(256), 0, stream>>>(xr, xi, mr, mi, out);
#else
  zgate_valu_kernel<<<dim3(ZN / 256), dim3(256), 0, stream>>>(xr, xi, mr, mi, out);
#endif
}